// NewCGCNReg_1563368096538
// MI455X (gfx1250) — compile-verified
//
#include <hip/hip_runtime.h>

// ---------------------------------------------------------------------------
// 2-layer GCN (PyG GCNConv semantics) for MI455X / gfx1250.
// N = 50000 nodes, E = 800000 edges, D = 128 features, hidden = 128, out = 1.
//
// Perf model (MI455X):
//  - GEMM x@W1 = 1.64 GFLOP -> trivial; done with native-precision
//    V_WMMA_F32_16X16X4_F32 (exact f32, matches reference numerics).
//    Operands staged in LDS via GLOBAL_LOAD_ASYNC_TO_LDS_B128 (ASYNCcnt
//    tracked, no VGPR round-trip).
//  - Edge aggregation = 102M gathered f32 (~410 MB) + 102M f32 atomics.
//    h1 (25.6MB) + accumulator (25.6MB) + edges fit in 192MB L2 -> phase is
//    L2 gather/atomic bound. One wave per edge, float4 per lane, hardware
//    global_atomic_add_f32.
// ---------------------------------------------------------------------------

#define DFEAT 128

typedef __attribute__((ext_vector_type(2))) float v2f;
typedef __attribute__((ext_vector_type(8))) float v8f;

#if defined(__has_builtin)
#if __has_builtin(__builtin_amdgcn_global_load_async_to_lds_b128)
#define USE_ASYNC_LDS 1
#endif
#endif

#ifdef USE_ASYNC_LDS
// Exact parameter types per hipcc diagnostic:
//   param0: int __attribute__((vector_size(16))) addrspace(1)*   (global src)
//   param1: int __attribute__((vector_size(16))) addrspace(3)*   (LDS dst)
//   param2/3: constant int offset / cpol
typedef int v4i_vs __attribute__((vector_size(16)));
typedef __attribute__((address_space(1))) v4i_vs glob_v4i;
typedef __attribute__((address_space(3))) v4i_vs lds_v4i;

__device__ __forceinline__ void async_copy16(const void* g, void* l) {
  // 16 bytes per lane, memory -> LDS, no VGPR data path; tracked by ASYNCcnt.
  __builtin_amdgcn_global_load_async_to_lds_b128(
      (glob_v4i*)g, (lds_v4i*)l, /*offset=*/0, /*cpol=*/0);
}

__device__ __forceinline__ void wait_async0() {
#if __has_builtin(__builtin_amdgcn_s_wait_asynccnt)
  __builtin_amdgcn_s_wait_asynccnt(0);
#else
  asm volatile("s_wait_asynccnt 0" ::: "memory");
#endif
}
#endif  // USE_ASYNC_LDS

__device__ __forceinline__ void atomic_fadd(float* p, float v) {
#if defined(__HIP_DEVICE_COMPILE__)
  unsafeAtomicAdd(p, v);   // lowers to global_atomic_add_f32
#else
  atomicAdd(p, v);
#endif
}

// ---- K1: zero the layer-1 accumulator, init deg to 1.0 (self loop) --------
__global__ void k_init(float* __restrict__ deg, float* __restrict__ agg, int N) {
  size_t i = (size_t)blockIdx.x * blockDim.x + threadIdx.x;
  size_t tot = (size_t)N * DFEAT;
  if (i < tot) agg[i] = 0.0f;
  if (i < (size_t)N) deg[i] = 1.0f;
}

// ---- K2: in-degree via atomics --------------------------------------------
__global__ void k_degree(const int* __restrict__ ei, float* __restrict__ deg, int E) {
  int e = blockIdx.x * blockDim.x + threadIdx.x;
  if (e < E) atomic_fadd(&deg[ei[E + e]], 1.0f);
}

// ---- K3: deg -> rsqrt(deg) in place (deg >= 1 always) ---------------------
__global__ void k_rsqrt(float* __restrict__ deg, int N) {
  int i = blockIdx.x * blockDim.x + threadIdx.x;
  if (i < N) deg[i] = rsqrtf(deg[i]);
}

// ---- K4: h1 = x @ W1 via V_WMMA_F32_16X16X4_F32 ---------------------------
// One block = 16 rows x 128 cols. 8 waves, one 16x16 tile each.
// A-strip (16x128, 8KB) and W1 (128x128, 64KB) staged in LDS (72KB of 320KB),
// via async global->LDS copies when available.
__global__ __launch_bounds__(256) void k_gemm_wmma(const float* __restrict__ X,
                                                   const float* __restrict__ W,
                                                   float* __restrict__ H) {
  __shared__ float sA[16 * DFEAT];
  __shared__ float sW[DFEAT * DFEAT];
  const int tid = threadIdx.x;
  const int row0 = blockIdx.x * 16;   // N_NODES is a multiple of 16

  const float4* __restrict__ Wv = (const float4*)W;
  float4* sWv = (float4*)sW;
  const float4* __restrict__ Xv = (const float4*)(X + (size_t)row0 * DFEAT);
  float4* sAv = (float4*)sA;

#ifdef USE_ASYNC_LDS
  // Direct memory -> LDS DMA, 16B per lane per issue; 16+2 issues per thread.
#pragma unroll 4
  for (int i = tid; i < (DFEAT * DFEAT) / 4; i += 256) async_copy16(Wv + i, sWv + i);
#pragma unroll
  for (int i = tid; i < (16 * DFEAT) / 4; i += 256) async_copy16(Xv + i, sAv + i);
  wait_async0();          // my wave's async copies have landed in LDS
  __syncthreads();        // everyone's copies visible to all waves
#else
#pragma unroll 4
  for (int i = tid; i < (DFEAT * DFEAT) / 4; i += 256) sWv[i] = Wv[i];
#pragma unroll
  for (int i = tid; i < (16 * DFEAT) / 4; i += 256) sAv[i] = Xv[i];
  __syncthreads();
#endif

  const int wave = tid >> 5;
  const int lane = tid & 31;
  const int col0 = wave * 16;
  const int mn = lane & 15;           // row of A-tile / col of B-tile
  const int kh = (lane >> 4) * 2;     // K sub-offset: lanes 0-15 -> K0/K1, 16-31 -> K2/K3

  v8f acc = {};
#pragma unroll
  for (int kk = 0; kk < DFEAT; kk += 4) {
    const int k0 = kk + kh;
    // A 16x4 f32 fragment: per-lane contiguous K pair -> ds_load_b64
    v2f a = *(const v2f*)(&sA[mn * DFEAT + k0]);
    // B 4x16 f32 fragment: B[k][n] = W1[k*128 + col0 + n]
    v2f b;
    b.x = sW[k0 * DFEAT + col0 + mn];
    b.y = sW[(k0 + 1) * DFEAT + col0 + mn];
    // (neg_a, A, neg_b, B, c_mod, C, reuse_a, reuse_b)
    acc = __builtin_amdgcn_wmma_f32_16x16x4_f32(false, a, false, b,
                                                (short)0, acc, false, false);
  }

  // C/D layout: VGPR v -> (M = v, N = lane) lanes 0-15 | (M = v+8, N = lane-16)
  const int mbase = row0 + (lane >> 4) * 8;
#pragma unroll
  for (int v = 0; v < 8; ++v)
    H[(size_t)(mbase + v) * DFEAT + col0 + mn] = acc[v];
}

// ---- K5: layer-1 message scatter. One wave per edge, float4 per lane ------
__global__ __launch_bounds__(256) void k_scatter1(const int* __restrict__ ei,
                                                  const float* __restrict__ dinv,
                                                  const float* __restrict__ h1,
                                                  float* __restrict__ agg, int E) {
  size_t t = (size_t)blockIdx.x * blockDim.x + threadIdx.x;
  size_t e = t >> 5;
  if (e >= (size_t)E) return;
  const int lane = (int)(t & 31);
  const int src = ei[e];
  const int dst = ei[(size_t)E + e];
  const float norm = dinv[src] * dinv[dst];
  const float4 v = *(const float4*)(h1 + (size_t)src * DFEAT + lane * 4);
  float* o = agg + (size_t)dst * DFEAT + lane * 4;
  atomic_fadd(o + 0, v.x * norm);
  atomic_fadd(o + 1, v.y * norm);
  atomic_fadd(o + 2, v.z * norm);
  atomic_fadd(o + 3, v.w * norm);
}

// ---- K6: self-loop + bias + ReLU, fused with s = h_relu @ W2 --------------
// One wave32 per node; 4 features per lane; shuffle tree reduction.
// Also initializes d_out with the layer-2 self-loop term + b2.
__global__ __launch_bounds__(256) void k_finalize(const float* __restrict__ agg,
                                                  const float* __restrict__ h1,
                                                  const float* __restrict__ dinv,
                                                  const float* __restrict__ b1,
                                                  const float* __restrict__ W2,
                                                  const float* __restrict__ b2,
                                                  float* __restrict__ s,
                                                  float* __restrict__ out, int N) {
  const int node = blockIdx.x * 8 + (threadIdx.x >> 5);
  const int lane = threadIdx.x & 31;
  if (node >= N) return;
  const float di = dinv[node];
  const float self = di * di;                       // norm of the self loop
  const float4 a = *(const float4*)(agg + (size_t)node * DFEAT + lane * 4);
  const float4 h = *(const float4*)(h1 + (size_t)node * DFEAT + lane * 4);
  const float4 bb = *(const float4*)(b1 + lane * 4);
  const float4 w = *(const float4*)(W2 + lane * 4);
  float4 r;
  r.x = fmaxf(fmaf(self, h.x, a.x) + bb.x, 0.0f);
  r.y = fmaxf(fmaf(self, h.y, a.y) + bb.y, 0.0f);
  r.z = fmaxf(fmaf(self, h.z, a.z) + bb.z, 0.0f);
  r.w = fmaxf(fmaf(self, h.w, a.w) + bb.w, 0.0f);
  float p = r.x * w.x + r.y * w.y + r.z * w.z + r.w * w.w;
#pragma unroll
  for (int off = 16; off > 0; off >>= 1) p += __shfl_xor(p, off, 32);
  if (lane == 0) {
    s[node] = p;
    out[node] = p * self + b2[0];                   // layer-2 self loop + bias
  }
}

// ---- K7: layer-2 scalar scatter -------------------------------------------
__global__ void k_scatter2(const int* __restrict__ ei, const float* __restrict__ dinv,
                           const float* __restrict__ s, float* __restrict__ out, int E) {
  int e = blockIdx.x * blockDim.x + threadIdx.x;
  if (e < E) {
    const int src = ei[e];
    const int dst = ei[E + e];
    atomic_fadd(&out[dst], s[src] * dinv[src] * dinv[dst]);
  }
}

extern "C" void kernel_launch(void* const* d_in, const int* in_sizes, int n_in,
                              void* d_out, int out_size, void* d_ws, size_t ws_size,
                              hipStream_t stream) {
  const float* x  = (const float*)d_in[0];
  const int*   ei = (const int*)d_in[1];
  const float* W1 = (const float*)d_in[2];
  const float* b1 = (const float*)d_in[3];
  const float* W2 = (const float*)d_in[4];
  const float* b2 = (const float*)d_in[5];
  float* out = (float*)d_out;

  const int N = in_sizes[0] / DFEAT;     // 50000
  const int E = in_sizes[1] / 2;         // 800000

  // Workspace carve-up (~51.7 MB): deg/dinv | h1 | agg | s
  float* deg = (float*)d_ws;
  float* h1  = deg + N;
  float* agg = h1 + (size_t)N * DFEAT;
  float* s   = agg + (size_t)N * DFEAT;

  const int B = 256;
  const size_t totInit = (size_t)N * DFEAT;

  k_init<<<(unsigned)((totInit + B - 1) / B), B, 0, stream>>>(deg, agg, N);
  k_degree<<<(E + B - 1) / B, B, 0, stream>>>(ei, deg, E);
  k_rsqrt<<<(N + B - 1) / B, B, 0, stream>>>(deg, N);
  k_gemm_wmma<<<N / 16, B, 0, stream>>>(x, W1, h1);
  {
    const size_t threads = (size_t)E * 32;
    k_scatter1<<<(unsigned)((threads + B - 1) / B), B, 0, stream>>>(ei, deg, h1, agg, E);
  }
  k_finalize<<<(N + 7) / 8, B, 0, stream>>>(agg, h1, deg, b1, W2, b2, s, out, N);
  k_scatter2<<<(E + B - 1) / B, B, 0, stream>>>(ei, deg, s, out, E);
}